// CriticWithMinibatch_81080392614177
// MI455X (gfx1250) — compile-verified
//
#include <hip/hip_runtime.h>
#include <hip/hip_bf16.h>

typedef __attribute__((ext_vector_type(2))) float v2f;
typedef __attribute__((ext_vector_type(8))) float v8f;

#define LEAKY_SLOPE 0.2f

__device__ __forceinline__ float lrelu(float v) { return v >= 0.0f ? v : LEAKY_SLOPE * v; }

// ---------------------------------------------------------------------------
// CDNA5 async global->LDS copy (ISA 15.18.3 op 96, tracked by ASYNCcnt).
// LDS destination address: per the flat-aperture rule (ISA 10.2), a generic
// pointer into LDS carries the LDS byte offset in addr[31:0], which is what
// the instruction's VDST operand wants.
// ---------------------------------------------------------------------------
__device__ __forceinline__ void async_load_f32_to_lds(const float* gsrc, float* ldst)
{
    unsigned int lds_off = (unsigned int)(unsigned long long)ldst;
    unsigned long long ga = (unsigned long long)gsrc;
    asm volatile("global_load_async_to_lds_b32 %0, %1, off"
                 :
                 : "v"(lds_off), "v"(ga)
                 : "memory");
}

__device__ __forceinline__ void async_wait_all()
{
    asm volatile("s_wait_asynccnt 0x0" ::: "memory");
}

// ---------------------------------------------------------------------------
// Generic fp32 GEMM via V_WMMA_F32_16X16X4_F32.
//   C = act(A @ B + bias)   A: MxK row-major, B: KxN row-major, C: MxN
// Block = 256 threads = 8 waves (wave32). Waves arranged 4(M) x 2(N).
// Each wave computes a 16x64 strip: one A fragment reused across 4 B tiles.
// Block tile: 64(M) x 128(N). All launch dims are exact multiples.
//
// VGPR layouts (ISA 7.12.2, 32-bit A 16x4 / B 4x16 / C,D 16x16):
//   A: lane m = lane&15 holds row m; element v holds K = 2*(lane>>4) + v
//   B: lane n = lane&15 holds col n; element v holds K = 2*(lane>>4) + v
//   D: element r, lanes 0-15 -> row r, lanes 16-31 -> row r+8; col = lane&15
// ---------------------------------------------------------------------------
template <bool RELU, bool BIAS>
__global__ __launch_bounds__(256) void gemm_wmma_f32(
    const float* __restrict__ A, const float* __restrict__ B,
    const float* __restrict__ bias, float* __restrict__ C,
    int N, int K)
{
    const int lane  = threadIdx.x & 31;
    const int wave  = threadIdx.x >> 5;
    const int waveM = wave & 3;   // 4 waves along M
    const int waveN = wave >> 2;  // 2 waves along N
    const int m     = lane & 15;
    const int g     = lane >> 4;  // lane group (0 or 1)

    const int rowBase = blockIdx.y * 64 + waveM * 16;
    const int colBase = blockIdx.x * 128 + waveN * 64;
    const int row     = rowBase + m;

    v8f acc[4] = {};

    const float* Arow = A + (size_t)row * K + 2 * g;  // even offset -> 8B aligned
    for (int k = 0; k < K; k += 4) {
        // stream A ahead (~1KB): lowers to global_prefetch_b8
        __builtin_prefetch(Arow + k + 256, 0, 1);
        v2f a = *(const v2f*)(Arow + k);
        const int kb = k + 2 * g;
#pragma unroll
        for (int t = 0; t < 4; ++t) {
            const int n = colBase + t * 16 + m;
            v2f b;
            b.x = B[(size_t)kb * N + n];
            b.y = B[(size_t)(kb + 1) * N + n];
            acc[t] = __builtin_amdgcn_wmma_f32_16x16x4_f32(
                /*neg_a=*/false, a, /*neg_b=*/false, b,
                /*c_mod=*/(short)0, acc[t],
                /*reuse_a=*/false, /*reuse_b=*/false);
        }
    }

#pragma unroll
    for (int t = 0; t < 4; ++t) {
        const int col = colBase + t * 16 + m;
        const float bv = BIAS ? bias[col] : 0.0f;
#pragma unroll
        for (int r = 0; r < 8; ++r) {
            const int orow = rowBase + r + 8 * g;
            float v = acc[t][r] + bv;
            if (RELU) v = lrelu(v);
            C[(size_t)orow * N + col] = v;
        }
    }
}

// ---------------------------------------------------------------------------
// Minibatch discrimination pairwise kernel.
//   Mb: (512, 128, 5) row-major (= GEMM3 output, col index = o*5+k)
//   ob[i,o] = sum_j exp(-sum_k |Mb[i,o,k]-Mb[j,o,k]|) - 1
// Block: 256 threads, tile = 32 i-rows x 16 o-cols; j tiles (32 x 80 f32,
// 10 KB) staged straight into LDS with async global->LDS loads (no VGPR
// bounce), completion via s_wait_asynccnt + barrier.
// ---------------------------------------------------------------------------
#define BN   512
#define OUTF 128
#define KD   5

__global__ __launch_bounds__(256) void minibatch_pairwise(
    const float* __restrict__ Mb, float* __restrict__ ob)
{
    __shared__ float smem[32 * 80];

    const int iBase = blockIdx.x * 32;
    const int oBase = blockIdx.y * 16;

    const int idx0 = threadIdx.x;
    const int il0 = idx0 >> 4, ol0 = idx0 & 15;
    const int idx1 = threadIdx.x + 256;
    const int il1 = idx1 >> 4, ol1 = idx1 & 15;

    float mi[2][KD];
    float acc0 = 0.0f, acc1 = 0.0f;
#pragma unroll
    for (int k = 0; k < KD; ++k) {
        mi[0][k] = Mb[(size_t)(iBase + il0) * (OUTF * KD) + (oBase + ol0) * KD + k];
        mi[1][k] = Mb[(size_t)(iBase + il1) * (OUTF * KD) + (oBase + ol1) * KD + k];
    }

    for (int j0 = 0; j0 < BN; j0 += 32) {
        __syncthreads();  // previous tile fully consumed before overwrite
#pragma unroll
        for (int e0 = 0; e0 < 32 * 80; e0 += 256) {
            const int e = e0 + threadIdx.x;
            const int jr = e / 80, c = e % 80;
            async_load_f32_to_lds(
                &Mb[(size_t)(j0 + jr) * (OUTF * KD) + oBase * KD + c],
                &smem[e]);
        }
        async_wait_all();
        __syncthreads();
#pragma unroll 4
        for (int jj = 0; jj < 32; ++jj) {
            const float* mj0 = &smem[jj * 80 + ol0 * KD];
            const float* mj1 = &smem[jj * 80 + ol1 * KD];
            float n0 = 0.0f, n1 = 0.0f;
#pragma unroll
            for (int k = 0; k < KD; ++k) {
                n0 += fabsf(mi[0][k] - mj0[k]);
                n1 += fabsf(mi[1][k] - mj1[k]);
            }
            acc0 += __expf(-n0);
            acc1 += __expf(-n1);
        }
    }

    ob[(size_t)(iBase + il0) * OUTF + oBase + ol0] = acc0 - 1.0f;
    ob[(size_t)(iBase + il1) * OUTF + oBase + ol1] = acc1 - 1.0f;
}

// z = concat([h2 (512x256), ob (512x128)], axis=1) -> (512, 384)
__global__ __launch_bounds__(256) void concat_kernel(
    const float* __restrict__ h2, const float* __restrict__ ob,
    float* __restrict__ z)
{
    const int i = blockIdx.x * blockDim.x + threadIdx.x;
    if (i >= 512 * 384) return;
    const int r = i / 384, c = i % 384;
    z[i] = (c < 256) ? h2[r * 256 + c] : ob[r * 128 + (c - 256)];
}

// out[r] = z3[r,:] @ W4 + b4   (512x128 @ 128x1)
__global__ __launch_bounds__(256) void final_head(
    const float* __restrict__ z3, const float* __restrict__ W4,
    const float* __restrict__ b4, float* __restrict__ out)
{
    const int r = blockIdx.x * blockDim.x + threadIdx.x;
    if (r >= 512) return;
    float s = 0.0f;
#pragma unroll 8
    for (int n = 0; n < 128; ++n) s += z3[(size_t)r * 128 + n] * W4[n];
    out[r] = s + b4[0];
}

extern "C" void kernel_launch(void* const* d_in, const int* in_sizes, int n_in,
                              void* d_out, int out_size, void* d_ws, size_t ws_size,
                              hipStream_t stream)
{
    (void)in_sizes; (void)n_in; (void)out_size; (void)ws_size;
    const float* x  = (const float*)d_in[0];
    const float* W1 = (const float*)d_in[1];
    const float* b1 = (const float*)d_in[2];
    const float* W2 = (const float*)d_in[3];
    const float* b2 = (const float*)d_in[4];
    const float* T  = (const float*)d_in[5];   // (256,128,5) == (256,640) row-major
    const float* W3 = (const float*)d_in[6];
    const float* b3 = (const float*)d_in[7];
    const float* W4 = (const float*)d_in[8];
    const float* b4 = (const float*)d_in[9];
    float* out = (float*)d_out;

    float* ws = (float*)d_ws;
    float* h1 = ws;                 // 512*512
    float* h2 = h1 + 512 * 512;     // 512*256
    float* Mb = h2 + 512 * 256;     // 512*640
    float* ob = Mb + 512 * 640;     // 512*128
    float* z  = ob + 512 * 128;     // 512*384
    float* z3 = z  + 512 * 384;     // 512*128

    const dim3 blk(256);
    // h1 = lrelu(x @ W1 + b1)        (512 x 512, K=1024)
    gemm_wmma_f32<true, true><<<dim3(512 / 128, 512 / 64), blk, 0, stream>>>(x, W1, b1, h1, 512, 1024);
    // h2 = lrelu(h1 @ W2 + b2)       (512 x 256, K=512)
    gemm_wmma_f32<true, true><<<dim3(256 / 128, 512 / 64), blk, 0, stream>>>(h1, W2, b2, h2, 256, 512);
    // Mb = h2 @ T                    (512 x 640, K=256)
    gemm_wmma_f32<false, false><<<dim3(640 / 128, 512 / 64), blk, 0, stream>>>(h2, T, nullptr, Mb, 640, 256);
    // ob = pairwise exp(-L1) sums    (512 x 128)
    minibatch_pairwise<<<dim3(512 / 32, 128 / 16), blk, 0, stream>>>(Mb, ob);
    // z = concat(h2, ob)             (512 x 384)
    concat_kernel<<<(512 * 384 + 255) / 256, blk, 0, stream>>>(h2, ob, z);
    // z3 = lrelu(z @ W3 + b3)        (512 x 128, K=384)
    gemm_wmma_f32<true, true><<<dim3(128 / 128, 512 / 64), blk, 0, stream>>>(z, W3, b3, z3, 128, 384);
    // out = z3 @ W4 + b4             (512 x 1)
    final_head<<<2, blk, 0, stream>>>(z3, W4, b4, out);
}